// HardTripletLoss2_20418274525812
// MI455X (gfx1250) — compile-verified
//
#include <hip/hip_runtime.h>
#include <hip/hip_bf16.h>
#include <math.h>

// HardTripletLoss2 for MI455X (gfx1250).
// Phase 1 (bandwidth-bound, ~511 MB @ 23.3 TB/s ~= 22us floor):
//   per-row  ||emb - attr + 1e-6||_2  using V_WMMA_F32_16X16X4_F32 with an
//   all-ones B matrix as a 16-row sum-reduction engine (matrix pipe does the
//   K-reduction, VALU only squares diffs). 16B-aligned global_load_b128.
// Phase 2 (tiny): column-wise hardest positive/negative + scalar loss.

typedef float v2f __attribute__((ext_vector_type(2)));
typedef float v8f __attribute__((ext_vector_type(8)));

#define D_DIM 312   // attribute dim (reference: D = 312, stride 1248B, 16B aligned)

__global__ void __launch_bounds__(256)
relations_wmma_kernel(const float* __restrict__ attr,
                      const float* __restrict__ emb,
                      float* __restrict__ rel,
                      int nRows)
{
    const int lane = threadIdx.x & 31;
    const int waveInBlk = threadIdx.x >> 5;
    const int rowBase = (blockIdx.x * (blockDim.x >> 5) + waveInBlk) << 4; // 16 rows/wave
    if (rowBase + 16 > nRows) return;            // uniform per wave: EXEC stays all-ones

    const int m  = lane & 15;                    // row within the 16-row tile
    const int hi = lane >> 4;                    // 0: K-lo half, 1: K-hi half
    const long base = (long)(rowBase + m) * D_DIM + hi * 4;

    v8f acc = {0.f, 0.f, 0.f, 0.f, 0.f, 0.f, 0.f, 0.f};
    const v2f bOnes = {1.0f, 1.0f};              // all-ones B => D[m,n] = sum_k A[m,k]

    for (int d = 0; d < D_DIM; d += 8) {
        const float4 e = *(const float4*)(emb  + base + d);
        const float4 a = *(const float4*)(attr + base + d);

        float qx = (e.x - a.x) + 1e-6f;
        float qy = (e.y - a.y) + 1e-6f;
        float qz = (e.z - a.z) + 1e-6f;
        float qw = (e.w - a.w) + 1e-6f;
        float sx = qx * qx, sy = qy * qy, sz = qz * qz, sw = qw * qw;

        // partner lane (xor 16) holds the other half of each 8-wide K chunk
        float px = __shfl_xor(sx, 16, 32);
        float py = __shfl_xor(sy, 16, 32);
        float pz = __shfl_xor(sz, 16, 32);
        float pw = __shfl_xor(sw, 16, 32);

        // A 16x4 f32 layout: lanes 0-15 -> (K0,K1) in (V0,V1); lanes 16-31 -> (K2,K3)
        v2f a1, a2;
        a1.x = hi ? pz : sx;   a1.y = hi ? pw : sy;   // K = d .. d+3
        a2.x = hi ? sz : px;   a2.y = hi ? sw : py;   // K = d+4 .. d+7

        acc = __builtin_amdgcn_wmma_f32_16x16x4_f32(false, a1, false, bOnes,
                                                    (short)0, acc, false, false);
        acc = __builtin_amdgcn_wmma_f32_16x16x4_f32(false, a2, false, bOnes,
                                                    (short)0, acc, false, false);
    }

    // D layout: VGPR r holds M=r (lanes 0-15, N=lane) and M=8+r (lanes 16-31, N=lane-16).
    // Column N=0 lives in lane 0 (rows 0-7) and lane 16 (rows 8-15).
    if (lane == 0) {
        #pragma unroll
        for (int r = 0; r < 8; ++r) rel[rowBase + r] = sqrtf(acc[r]);
    } else if (lane == 16) {
        #pragma unroll
        for (int r = 0; r < 8; ++r) rel[rowBase + 8 + r] = sqrtf(acc[r]);
    }
}

__global__ void __launch_bounds__(256)
hard_triplet_loss_kernel(const float* __restrict__ rel,
                         const int* __restrict__ labels,
                         float* __restrict__ out,
                         int Bn, int Cn)
{
    __shared__ int   sLab[1024];
    __shared__ float sSum[256];
    __shared__ float sCnt[256];

    for (int i = threadIdx.x; i < Bn; i += blockDim.x) sLab[i] = labels[i];
    __syncthreads();

    const int c = threadIdx.x;
    float tl = 0.0f;
    if (c < Cn) {
        float maxAll = -INFINITY;    // column max over all anchors
        float hp     = -INFINITY;    // max of relations * mask_pos
        float minU   =  INFINITY;    // min over unmasked rows
        float minM   =  INFINITY;    // min over masked rows
        for (int b = 0; b < Bn; ++b) {
            float r = rel[b * Cn + c];
            bool  msk = (sLab[b] == c);
            maxAll = fmaxf(maxAll, r);
            hp     = fmaxf(hp, msk ? r : 0.0f);
            if (msk) minM = fminf(minM, r);
            else     minU = fminf(minU, r);
        }
        // anchor_negative = rel + maxAll*mask; hardest_negative = min over rows
        float hn = fminf(minU, minM + maxAll);   // minM=+inf when class unused -> minU
        tl = hp - hn + 1.0f;                     // MARGIN = 1.0
        tl = tl > 0.0f ? tl : 0.0f;              // relu
    }
    float cnt = (tl > 1e-16f) ? 1.0f : 0.0f;

    sSum[threadIdx.x] = tl;
    sCnt[threadIdx.x] = cnt;
    __syncthreads();
    for (int s = 128; s > 0; s >>= 1) {
        if ((int)threadIdx.x < s) {
            sSum[threadIdx.x] += sSum[threadIdx.x + s];
            sCnt[threadIdx.x] += sCnt[threadIdx.x + s];
        }
        __syncthreads();
    }
    if (threadIdx.x == 0) out[0] = sSum[0] / (sCnt[0] + 1e-16f);
}

extern "C" void kernel_launch(void* const* d_in, const int* in_sizes, int n_in,
                              void* d_out, int out_size, void* d_ws, size_t ws_size,
                              hipStream_t stream)
{
    const float* attr   = (const float*)d_in[0];
    const float* emb    = (const float*)d_in[1];
    const int*   labels = (const int*)d_in[2];
    float*       out    = (float*)d_out;

    const int Bn    = in_sizes[2];            // 1024 anchors
    const int total = in_sizes[0];            // B*C*D
    const int nRows = total / D_DIM;          // B*C = 204800
    const int Cn    = nRows / Bn;             // 200 classes

    float* rel = (float*)d_ws;                // (B*C) floats of scratch

    // 8 waves/block * 16 rows/wave = 128 rows per block
    const int rowsPerBlock = 128;
    const int grid = (nRows + rowsPerBlock - 1) / rowsPerBlock;   // 1600 blocks

    relations_wmma_kernel<<<grid, 256, 0, stream>>>(attr, emb, rel, nRows);
    hard_triplet_loss_kernel<<<1, 256, 0, stream>>>(rel, labels, out, Bn, Cn);
}